// BatchedLensBank_39024072852175
// MI455X (gfx1250) — compile-verified
//
#include <hip/hip_runtime.h>
#include <hip/hip_bf16.h>
#include <math.h>
#include <stdint.h>

// CDNA5 / gfx1250, wave32.
typedef float v2f  __attribute__((ext_vector_type(2)));
typedef float v8f  __attribute__((ext_vector_type(8)));
typedef unsigned int u32x4 __attribute__((ext_vector_type(4)));
typedef int   i32x4 __attribute__((ext_vector_type(4)));
typedef int   i32x8 __attribute__((ext_vector_type(8)));

#define LN_EPS 1e-5f
#define N_BANK 256
#define D_IN   4096
#define H1_DIM 256
#define H2_DIM 64
#define BLOCK  512                 // 16 wave32s
#define W2_PAD 258                 // 256 + 2 floats: LDS bank-conflict padding

#if __has_builtin(__builtin_amdgcn_tensor_load_to_lds)
#define USE_TDM 1
#else
#define USE_TDM 0
#endif

__device__ __forceinline__ float wave_reduce_add(float v) {
    // wave32 butterfly reduction
    #pragma unroll
    for (int off = 16; off > 0; off >>= 1)
        v += __shfl_xor(v, off, 32);
    return v;
}

__global__ __launch_bounds__(BLOCK)
void BatchedLensBank_39024072852175_kernel(
    const float* __restrict__ x,      // (D)
    const float* __restrict__ LN_w,   // (N, D)
    const float* __restrict__ LN_b,   // (N, D)
    const float* __restrict__ W1,     // (N, H1, D)
    const float* __restrict__ b1,     // (N, H1)
    const float* __restrict__ W2,     // (N, H2, H1)
    const float* __restrict__ b2,     // (N, H2)
    const float* __restrict__ W3,     // (N, 1, H2)
    const float* __restrict__ b3,     // (N, 1)
    float* __restrict__ out)          // probs(N) ++ logits(N)
{
    const int n    = blockIdx.x;            // one bank member per workgroup
    const int tid  = threadIdx.x;
    const int wave = tid >> 5;              // 0..15
    const int lane = tid & 31;              // wave32

    __shared__ __align__(16) float xe_s[D_IN];                 // 16 KB
    __shared__ __align__(16) float w2_s[H2_DIM * W2_PAD];      // ~64.5 KB (padded rows)
    __shared__ __align__(16) float h1_s[H1_DIM];
    __shared__ __align__(16) float h2_s[H2_DIM];
    __shared__ float red_s[32];                                // [0..15] sum, [16..31] sumsq
    __shared__ float stats_s[2];                               // mean, rstd

#if USE_TDM
    // ---------------- TDM: DMA W2[n] (64 x 256 f32) into LDS -------------------
    // Tensor Data Mover with hardware LDS padding: pad_interval = 256 DWORDs
    // (one row), pad_amount = 2 DWORDs -> LDS rows land at stride 258 floats,
    // exactly the W2_PAD layout the WMMA stage expects. Issued once by wave 0
    // at kernel start so the 64 KB DMA overlaps LayerNorm + the W1 stream.
    if (wave == 0) {
        const uint64_t ga  = (uint64_t)(uintptr_t)(W2 + (size_t)n * H2_DIM * H1_DIM);
        const uint32_t lds = (uint32_t)(uintptr_t)(&w2_s[0]);   // flat low 32 = LDS offset
        u32x4 g0;
        g0.x = 1u;                                              // count=1 (valid user D#)
        g0.y = lds;                                             // lds_addr (bytes)
        g0.z = (uint32_t)ga;                                    // global_addr[31:0]
        g0.w = ((uint32_t)(ga >> 32) & 0x01FFFFFFu)             // global_addr[56:32]
             | 0x80000000u;                                     // type=2 ("image")
        i32x8 g1;
        g1[0] = (2 << 16)        // data_size = 2 -> 4-byte elements
              | (1 << 20)        // pad_enable
              | (7 << 22)        // pad_interval: 256 DWORDs between pads
              | (1 << 25);       // pad_amount: 2 DWORDs of padding
        g1[1] = (int)(256u << 16);   // tensor_dim0[15:0] = 256 (row length, elems)
        g1[2] = (int)(64u  << 16);   // tensor_dim0 hi = 0 ; tensor_dim1[15:0] = 64
        g1[3] = (int)(256u << 16);   // tensor_dim1 hi = 0 ; tile_dim0 = 256
        g1[4] = 64;                  // tile_dim1 = 64 ; tile_dim2 = 0
        g1[5] = 256;                 // tensor_dim0_stride[31:0] = 256 elems
        g1[6] = 0;                   // stride0 hi, dim1_stride lo (unused, 2D)
        g1[7] = 0;
        const i32x4 gz4 = {0, 0, 0, 0};                         // groups 2/3: 2D tensor
        const i32x8 gz8 = {0, 0, 0, 0, 0, 0, 0, 0};
        // amdgpu-toolchain (clang-23) 6-arg form:
        // (g0 v4u32, g1 v8i32, g2 v4i32, g3 v4i32, v8i32, cpol)
        __builtin_amdgcn_tensor_load_to_lds(g0, g1, gz4, gz4, gz8, 0);
    }
#endif

    // ---------------- Stage 0: LayerNorm stats of shared x (block reduce) -------
    // Each thread owns 8 contiguous elements of x (wave covers 1 KB contiguous).
    const int d0 = tid * 8;
    float4 xa = *(const float4*)(x + d0);
    float4 xb = *(const float4*)(x + d0 + 4);
    float s1 = xa.x + xa.y + xa.z + xa.w + xb.x + xb.y + xb.z + xb.w;
    float s2 = xa.x*xa.x + xa.y*xa.y + xa.z*xa.z + xa.w*xa.w
             + xb.x*xb.x + xb.y*xb.y + xb.z*xb.z + xb.w*xb.w;
    s1 = wave_reduce_add(s1);
    s2 = wave_reduce_add(s2);
    if (lane == 0) { red_s[wave] = s1; red_s[16 + wave] = s2; }
    __syncthreads();
    if (tid == 0) {
        float ts1 = 0.f, ts2 = 0.f;
        #pragma unroll
        for (int i = 0; i < 16; ++i) { ts1 += red_s[i]; ts2 += red_s[16 + i]; }
        float mean = ts1 * (1.0f / D_IN);
        float var  = ts2 * (1.0f / D_IN) - mean * mean;
        stats_s[0] = mean;
        stats_s[1] = rsqrtf(var + LN_EPS);
    }
    __syncthreads();
    const float mean = stats_s[0];
    const float rstd = stats_s[1];

    // ---------------- Stage 1: xe = x_norm * LN_w[n] + LN_b[n]  -> LDS ----------
    {
        const size_t base = (size_t)n * D_IN + d0;
        float4 wa = *(const float4*)(LN_w + base);
        float4 wb = *(const float4*)(LN_w + base + 4);
        float4 ba = *(const float4*)(LN_b + base);
        float4 bb = *(const float4*)(LN_b + base + 4);
        float4 ea, eb;
        ea.x = (xa.x - mean) * rstd * wa.x + ba.x;
        ea.y = (xa.y - mean) * rstd * wa.y + ba.y;
        ea.z = (xa.z - mean) * rstd * wa.z + ba.z;
        ea.w = (xa.w - mean) * rstd * wa.w + ba.w;
        eb.x = (xb.x - mean) * rstd * wb.x + bb.x;
        eb.y = (xb.y - mean) * rstd * wb.y + bb.y;
        eb.z = (xb.z - mean) * rstd * wb.z + bb.z;
        eb.w = (xb.w - mean) * rstd * wb.w + bb.w;
        *(float4*)(xe_s + d0)     = ea;
        *(float4*)(xe_s + d0 + 4) = eb;
    }
#if !USE_TDM
    // Fallback: stage W2[n] into padded LDS with coalesced float4 loads.
    {
        const float4* w2g = (const float4*)(W2 + (size_t)n * H2_DIM * H1_DIM);
        #pragma unroll
        for (int v = 0; v < 8; ++v) {
            int idx  = tid + v * BLOCK;            // float4 index, 0..4095
            int flat = idx * 4;
            int row  = flat >> 8;                  // /256
            int col  = flat & 255;
            *(float4*)(w2_s + row * W2_PAD + col) = w2g[idx];
        }
    }
#endif
    __syncthreads();

    // ---------------- Stage 2: h1 = relu(W1[n] @ xe + b1[n]) --------------------
    // Bandwidth-critical: 1.07 GB of W1 streamed across the grid.
    // Each wave streams a full 4096-row with 32 lanes * float4 (512B/step).
    {
        const float4* xe4 = (const float4*)xe_s;
        #pragma unroll 2
        for (int j = 0; j < 16; ++j) {
            const int r = wave * 16 + j;
            const float4* wrow =
                (const float4*)(W1 + ((size_t)n * H1_DIM + r) * D_IN);
            float4 acc = {0.f, 0.f, 0.f, 0.f};
            for (int i = lane; i < D_IN / 4; i += 32) {
                __builtin_prefetch(&wrow[i + 64], 0, 0);   // global_prefetch_b8 lookahead
                float4 wv = wrow[i];
                float4 xv = xe4[i];
                acc.x = fmaf(wv.x, xv.x, acc.x);
                acc.y = fmaf(wv.y, xv.y, acc.y);
                acc.z = fmaf(wv.z, xv.z, acc.z);
                acc.w = fmaf(wv.w, xv.w, acc.w);
            }
            float sum = wave_reduce_add((acc.x + acc.y) + (acc.z + acc.w));
            if (lane == 0) {
                float v = sum + b1[(size_t)n * H1_DIM + r];
                h1_s[r] = v > 0.f ? v : 0.f;
            }
        }
    }
#if USE_TDM
    // Publish the TDM result: issuing wave drains TENSORcnt, then the barrier
    // makes w2_s visible to the WMMA waves.
    if (wave == 0) {
#if __has_builtin(__builtin_amdgcn_s_wait_tensorcnt)
        __builtin_amdgcn_s_wait_tensorcnt(0);
#else
        asm volatile("s_wait_tensorcnt 0x0" ::: "memory");
#endif
    }
#endif
    __syncthreads();

    // ---------------- Stage 3: h2 = relu(W2[n] @ h1 + b2[n]) via f32 WMMA -------
    // V_WMMA_F32_16X16X4_F32: A = 16x4 tile of W2 rows, B = h1[k..k+3] broadcast
    // across all 16 columns, so every D column carries the GEMV result.
    // Layouts per ISA 7.12.2: A/B VGPR0 holds K={0,2} halves, VGPR1 K={1,3};
    // C/D VGPR r: lanes 0-15 -> M=r, lanes 16-31 -> M=8+r.
    if (wave < 4) {
        const int rowBase = wave * 16;
        const int m    = lane & 15;
        const int half = lane >> 4;
        const float* arow = &w2_s[(rowBase + m) * W2_PAD];
        v8f c = {};
        #pragma unroll 4
        for (int k0 = 0; k0 < H1_DIM; k0 += 4) {
            const int ka = k0 + 2 * half;
            v2f a, b;
            a.x = arow[ka];
            a.y = arow[ka + 1];
            b.x = h1_s[ka];          // broadcast across columns (same addr per half)
            b.y = h1_s[ka + 1];
            c = __builtin_amdgcn_wmma_f32_16x16x4_f32(
                    /*neg_a=*/false, a, /*neg_b=*/false, b,
                    /*c_mod=*/(short)0, c, /*reuse_a=*/false, /*reuse_b=*/false);
        }
        // Every column identical: lane 0 carries M=0..7, lane 16 carries M=8..15.
        if (lane == 0 || lane == 16) {
            const int rOff = rowBase + (half ? 8 : 0);
            #pragma unroll
            for (int r = 0; r < 8; ++r) {
                float v = c[r] + b2[(size_t)n * H2_DIM + rOff + r];
                h2_s[rOff + r] = v > 0.f ? v : 0.f;
            }
        }
    }
    __syncthreads();

    // ---------------- Stage 4: logits / probs -----------------------------------
    if (wave == 0) {
        const int e = lane * 2;
        const float* w3 = W3 + (size_t)n * H2_DIM;
        float s = fmaf(w3[e], h2_s[e], w3[e + 1] * h2_s[e + 1]);
        s = wave_reduce_add(s);
        if (lane == 0) {
            float logit = s + b3[n];
            out[N_BANK + n] = logit;                       // logits
            out[n] = 1.0f / (1.0f + expf(-logit));         // probs
        }
    }
}

extern "C" void kernel_launch(void* const* d_in, const int* in_sizes, int n_in,
                              void* d_out, int out_size, void* d_ws, size_t ws_size,
                              hipStream_t stream) {
    (void)in_sizes; (void)n_in; (void)d_ws; (void)ws_size; (void)out_size;
    const float* x    = (const float*)d_in[0];
    const float* LN_w = (const float*)d_in[1];
    const float* LN_b = (const float*)d_in[2];
    const float* W1   = (const float*)d_in[3];
    const float* b1   = (const float*)d_in[4];
    const float* W2   = (const float*)d_in[5];
    const float* b2   = (const float*)d_in[6];
    const float* W3   = (const float*)d_in[7];
    const float* b3   = (const float*)d_in[8];
    float* out = (float*)d_out;

    BatchedLensBank_39024072852175_kernel<<<N_BANK, BLOCK, 0, stream>>>(
        x, LN_w, LN_b, W1, b1, W2, b2, W3, b3, out);
}